// DropHeteroGIN_24137716203678
// MI455X (gfx1250) — compile-verified
//
#include <hip/hip_runtime.h>

typedef __attribute__((ext_vector_type(2))) float v2f;
typedef __attribute__((ext_vector_type(4))) float v4f;
typedef __attribute__((ext_vector_type(8))) float v8f;

#define D_      128
#define RUNS_   4
#define NAUT    50000
#define NPAP    100000
#define NEDGE   500000
#define LDA_    132        // 128 + 4 pad: row stride coprime-ish with 64 banks
#define INV_KEEP 1.25f     // 1/(1-0.2)

// ---------------------------------------------------------------------------
// Fused WMMA fp32 GEMM: C[M,128] = A'[M,128] x W[128,128] + bias, with
// mode-dependent input transform (fused at LDS staging) and epilogue.
//   MODE 0 (PROJ) : A' = X                        ; out = acc + b
//   MODE 1 (PASS1): A' = (1+eps)*X + (row<Nbase ? AGG : 0)
//                   out = acc + b1 -> Out, and accumulate BN column sum/sumsq
//   MODE 2 (PASS2): A' = relu(X*scale + shift)    ; out = relu(acc + b2)
// Block: 128 threads = 4 waves; 64 rows x 128 cols per block.
// Each wave: one 16-row stripe, 8 col-tiles, 32 x V_WMMA_F32_16X16X4_F32.
// ---------------------------------------------------------------------------
template<int MODE>
__global__ __launch_bounds__(128) void gemm128_kernel(
    const float* __restrict__ X, const float* __restrict__ AGG, int Nbase,
    const float* __restrict__ epsp, const float* __restrict__ W,
    const float* __restrict__ bias, const float* __restrict__ scale,
    const float* __restrict__ shift, float* __restrict__ Out,
    float* __restrict__ stats, int M)
{
    __shared__ float lA[64 * LDA_];
    __shared__ float lred[256];

    const int tid  = threadIdx.x;
    const int wave = tid >> 5;
    const int lane = tid & 31;
    const int lr   = lane & 15;   // column-in-tile / row-in-Afrag
    const int lh   = lane >> 4;   // half selector
    const int rowbase = blockIdx.x * 64;

    if (MODE == 1) { lred[tid] = 0.0f; lred[tid + 128] = 0.0f; }

    float ep1 = 1.0f;
    if (MODE == 1) ep1 = 1.0f + epsp[0];

    // ---- stage A tile (64 x 128) into LDS, transform fused, float4 loads ----
    {
        const int rsub = tid >> 5;          // 4 rows per sweep
        const int kv   = (lane) << 2;       // float4 column offset
        #pragma unroll
        for (int it = 0; it < 16; ++it) {
            const int r   = it * 4 + rsub;
            const int row = rowbase + r;
            v4f a = (v4f){0.f, 0.f, 0.f, 0.f};
            if (row < M) {
                a = *(const v4f*)(X + (size_t)row * D_ + kv);
                if (MODE == 1) {
                    a.x *= ep1; a.y *= ep1; a.z *= ep1; a.w *= ep1;
                    if (row < Nbase) {
                        v4f g = *(const v4f*)(AGG + (size_t)row * D_ + kv);
                        a.x += g.x; a.y += g.y; a.z += g.z; a.w += g.w;
                    }
                } else if (MODE == 2) {
                    a.x = fmaxf(a.x * scale[kv + 0] + shift[kv + 0], 0.f);
                    a.y = fmaxf(a.y * scale[kv + 1] + shift[kv + 1], 0.f);
                    a.z = fmaxf(a.z * scale[kv + 2] + shift[kv + 2], 0.f);
                    a.w = fmaxf(a.w * scale[kv + 3] + shift[kv + 3], 0.f);
                }
            }
            *(v4f*)(&lA[r * LDA_ + kv]) = a;
        }
    }
    __syncthreads();

    // ---- WMMA main loop: K = 128 in steps of 4 ----
    const int r16 = wave * 16;
    v8f acc[8];
    #pragma unroll
    for (int c = 0; c < 8; ++c) acc[c] = (v8f){0,0,0,0,0,0,0,0};

    #pragma unroll 4
    for (int kb = 0; kb < 32; ++kb) {
        const int k0 = kb * 4 + lh * 2;
        // A frag (16x4 f32): lane half picks K pair, vgpr picks within pair
        v2f a;
        a.x = lA[(r16 + lr) * LDA_ + k0];
        a.y = lA[(r16 + lr) * LDA_ + k0 + 1];
        #pragma unroll
        for (int c = 0; c < 8; ++c) {
            const int n = c * 16 + lr;
            v2f b;                            // B frag (4x16 f32)
            b.x = W[(size_t)k0 * D_ + n];
            b.y = W[(size_t)(k0 + 1) * D_ + n];
            acc[c] = __builtin_amdgcn_wmma_f32_16x16x4_f32(
                false, a, false, b, (short)0, acc[c], false, false);
        }
    }

    // ---- epilogue: C/D layout row = v + 8*lh, col = 16c + lr ----
    #pragma unroll
    for (int c = 0; c < 8; ++c) {
        const int col = c * 16 + lr;
        const float bi = bias[col];
        float ssum = 0.f, ssq = 0.f;
        #pragma unroll
        for (int v = 0; v < 8; ++v) {
            const int row = rowbase + r16 + v + 8 * lh;
            if (row < M) {
                float val = acc[c][v] + bi;
                if (MODE == 2) val = fmaxf(val, 0.f);
                Out[(size_t)row * D_ + col] = val;
                if (MODE == 1) { ssum += val; ssq += val * val; }
            }
        }
        if (MODE == 1) {
            atomicAdd(&lred[col], ssum);
            atomicAdd(&lred[col + 128], ssq);
        }
    }
    if (MODE == 1) {
        __syncthreads();
        atomicAdd(&stats[tid], lred[tid]);
        atomicAdd(&stats[tid + 128], lred[tid + 128]);
    }
}

// Fold BN batch stats into per-column scale/shift: stats[256..] / stats[384..]
__global__ __launch_bounds__(128) void bnprep_kernel(
    float* __restrict__ stats, const float* __restrict__ g,
    const float* __restrict__ bt, float invM)
{
    const int t = threadIdx.x;
    const float mean = stats[t] * invM;
    const float var  = stats[128 + t] * invM - mean * mean;
    const float rstd = rsqrtf(var + 1e-5f);
    const float sc   = g[t] * rstd;
    stats[256 + t] = sc;
    stats[384 + t] = bt[t] - mean * sc;
}

// h[r,n,:] = proj[n,:] * mask[r,n] * inv_keep
__global__ __launch_bounds__(256) void expand_kernel(
    const float* __restrict__ proj, const float* __restrict__ mask,
    float* __restrict__ out, int N)
{
    const size_t idx = (size_t)blockIdx.x * 256 + threadIdx.x;
    const int d = (int)(idx & (D_ - 1));
    const size_t nd = idx >> 7;                 // r*N + n == mask index
    const int n = (int)(nd % (size_t)N);
    out[idx] = proj[(size_t)n * D_ + d] * mask[nd] * INV_KEEP;
}

// agg[dst,:] += X[src,:]   (run-0 rows only; one block per edge)
__global__ __launch_bounds__(128) void scatter_kernel(
    const float* __restrict__ X, const int* __restrict__ src,
    const int* __restrict__ dst, float* __restrict__ agg)
{
    const int e = blockIdx.x;
    const int s = src[e];
    const int d = dst[e];
    const int t = threadIdx.x;
    atomicAdd(&agg[(size_t)d * D_ + t], X[(size_t)s * D_ + t]);
}

// out[a,:] = mean_r(hA[r,a,:]) @ fw + fb     (4 classes)
__global__ __launch_bounds__(128) void final_kernel(
    const float* __restrict__ hA, const float* __restrict__ fw,
    const float* __restrict__ fb, float* __restrict__ out)
{
    __shared__ float red[4][128];
    const int a = blockIdx.x;
    const int t = threadIdx.x;
    float m = 0.f;
    #pragma unroll
    for (int r = 0; r < RUNS_; ++r)
        m += hA[((size_t)r * NAUT + a) * D_ + t];
    m *= 0.25f;
    #pragma unroll
    for (int c = 0; c < 4; ++c) red[c][t] = m * fw[t * 4 + c];
    __syncthreads();
    for (int s = 64; s > 0; s >>= 1) {
        if (t < s) {
            #pragma unroll
            for (int c = 0; c < 4; ++c) red[c][t] += red[c][t + s];
        }
        __syncthreads();
    }
    if (t < 4) out[(size_t)a * 4 + t] = red[t][0] + fb[t];
}

extern "C" void kernel_launch(void* const* d_in, const int* in_sizes, int n_in,
                              void* d_out, int out_size, void* d_ws, size_t ws_size,
                              hipStream_t stream) {
    (void)in_sizes; (void)n_in; (void)out_size; (void)ws_size;
    const float* x_author = (const float*)d_in[0];
    const float* x_paper  = (const float*)d_in[1];
    const int*   ei_ap    = (const int*)d_in[2];   // [2,E]: src(author), dst(paper)
    const int*   ei_pa    = (const int*)d_in[3];   // [2,E]: src(paper), dst(author)
    const float* drop_a   = (const float*)d_in[4];
    const float* drop_p   = (const float*)d_in[5];
    const float* lin_a_w  = (const float*)d_in[6];
    const float* lin_a_b  = (const float*)d_in[7];
    const float* lin_p_w  = (const float*)d_in[8];
    const float* lin_p_b  = (const float*)d_in[9];
    const float* W1 = (const float*)d_in[10];
    const float* B1 = (const float*)d_in[11];
    const float* G  = (const float*)d_in[12];
    const float* BT = (const float*)d_in[13];
    const float* W2 = (const float*)d_in[14];
    const float* B2 = (const float*)d_in[15];
    const float* EPS = (const float*)d_in[16];
    const float* fw = (const float*)d_in[17];
    const float* fb = (const float*)d_in[18];
    float* out = (float*)d_out;

    // workspace carve-up (floats)
    float* ws = (float*)d_ws;
    size_t off = 0;
    float* hA   = ws + off; off += (size_t)RUNS_ * NAUT * D_;   // 102.4 MB
    float* hP   = ws + off; off += (size_t)RUNS_ * NPAP * D_;   // 204.8 MB
    float* Hbuf = ws + off; off += (size_t)RUNS_ * NPAP * D_;   // 204.8 MB
    float* aggP = ws + off; off += (size_t)NPAP * D_;           // 51.2 MB
    float* aggA = ws + off; off += (size_t)NAUT * D_;           // 25.6 MB
    float* stats = ws + off;                                     // 2 KB region

    const int MA = RUNS_ * NAUT;   // 200000
    const int MP = RUNS_ * NPAP;   // 400000

    // --- input projections + DropGNN expand ---
    gemm128_kernel<0><<<(NAUT + 63) / 64, 128, 0, stream>>>(
        x_author, nullptr, 0, nullptr, lin_a_w, lin_a_b, nullptr, nullptr,
        Hbuf, nullptr, NAUT);
    expand_kernel<<<(int)(((size_t)MA * D_) / 256), 256, 0, stream>>>(
        Hbuf, drop_a, hA, NAUT);
    gemm128_kernel<0><<<(NPAP + 63) / 64, 128, 0, stream>>>(
        x_paper, nullptr, 0, nullptr, lin_p_w, lin_p_b, nullptr, nullptr,
        Hbuf, nullptr, NPAP);
    expand_kernel<<<(int)(((size_t)MP * D_) / 256), 256, 0, stream>>>(
        Hbuf, drop_p, hP, NPAP);

    // --- L = 2 GIN layers ---
    for (int l = 0; l < 2; ++l) {
        const int e0 = l * 2 + 0;   // author -> paper update
        const int e1 = l * 2 + 1;   // paper -> author update

        hipMemsetAsync(aggP, 0, (size_t)NPAP * D_ * sizeof(float), stream);
        hipMemsetAsync(aggA, 0, (size_t)NAUT * D_ * sizeof(float), stream);
        // both aggregates from PRE-update features
        scatter_kernel<<<NEDGE, 128, 0, stream>>>(hA, ei_ap, ei_ap + NEDGE, aggP);
        scatter_kernel<<<NEDGE, 128, 0, stream>>>(hP, ei_pa, ei_pa + NEDGE, aggA);

        // papers: z = (1+eps)hP + aggP ; h = z W1 + b1 ; BN ; relu ; W2+b2 ; relu
        hipMemsetAsync(stats, 0, 256 * sizeof(float), stream);
        gemm128_kernel<1><<<MP / 64, 128, 0, stream>>>(
            hP, aggP, NPAP, EPS + e0, W1 + (size_t)e0 * D_ * D_, B1 + (size_t)e0 * D_,
            nullptr, nullptr, Hbuf, stats, MP);
        bnprep_kernel<<<1, 128, 0, stream>>>(
            stats, G + (size_t)e0 * D_, BT + (size_t)e0 * D_, 1.0f / (float)MP);
        gemm128_kernel<2><<<MP / 64, 128, 0, stream>>>(
            Hbuf, nullptr, 0, nullptr, W2 + (size_t)e0 * D_ * D_, B2 + (size_t)e0 * D_,
            stats + 256, stats + 384, hP, nullptr, MP);

        // authors
        hipMemsetAsync(stats, 0, 256 * sizeof(float), stream);
        gemm128_kernel<1><<<MA / 64, 128, 0, stream>>>(
            hA, aggA, NAUT, EPS + e1, W1 + (size_t)e1 * D_ * D_, B1 + (size_t)e1 * D_,
            nullptr, nullptr, Hbuf, stats, MA);
        bnprep_kernel<<<1, 128, 0, stream>>>(
            stats, G + (size_t)e1 * D_, BT + (size_t)e1 * D_, 1.0f / (float)MA);
        gemm128_kernel<2><<<MA / 64, 128, 0, stream>>>(
            Hbuf, nullptr, 0, nullptr, W2 + (size_t)e1 * D_ * D_, B2 + (size_t)e1 * D_,
            stats + 256, stats + 384, hA, nullptr, MA);
    }

    // --- run-mean over authors + classifier head ---
    final_kernel<<<NAUT, 128, 0, stream>>>(hA, fw, fb, out);
}